// SymmetricContraction_58729382805671
// MI455X (gfx1250) — compile-verified
//
#include <hip/hip_runtime.h>

typedef float v2f __attribute__((ext_vector_type(2)));
typedef float v8f __attribute__((ext_vector_type(8)));

#define UCOLS 96      // 94 used columns (order1:9, order2:23, order3:62), padded
#define UROWS 820     // 819 monomials (9 + 81 + 729), padded to multiple of 4
#define NROW  131072  // N * C
#define TILE_ROWS 16
#define BLOCK_THREADS 192  // 6 waves: one per 16-column tile

// ---------------- Stage 0a: zero the transposed-U layout buffer ----------------
__global__ void zero_u(float* __restrict__ UT) {
  int t = blockIdx.x * blockDim.x + threadIdx.x;
  int stride = gridDim.x * blockDim.x;
  for (int i = t; i < UROWS * UCOLS; i += stride) UT[i] = 0.f;
}

// ------- Stage 0b: scatter u tensors into UT[col][row] (layout transform only) -
__device__ inline void scatter_u(const float* __restrict__ u, int nD, int rowbase,
                                 int mul, int idim, int colbase,
                                 float* __restrict__ UT, int t, int stride) {
  const int mi = mul * idim;
  const int total = nD * mi;
  for (int e = t; e < total; e += stride) {
    int dd  = e / mi;          // flat (d0[,d1[,d2]]) row-major == monomial (K) index
    int rem = e - dd * mi;     // k*idim + i == column offset within irrep block
    UT[(colbase + rem) * UROWS + rowbase + dd] = u[e];
  }
}

__global__ void fill_u(const float* __restrict__ u1_0e, const float* __restrict__ u1_1o,
                       const float* __restrict__ u1_2e, const float* __restrict__ u2_0e,
                       const float* __restrict__ u2_1o, const float* __restrict__ u2_2e,
                       const float* __restrict__ u3_0e, const float* __restrict__ u3_1o,
                       const float* __restrict__ u3_2e, float* __restrict__ UT) {
  int t = blockIdx.x * blockDim.x + threadIdx.x;
  int stride = gridDim.x * blockDim.x;
  // order 1: K rows 0..8, cols 0..8
  scatter_u(u1_0e,   9,  0, 1, 1,  0, UT, t, stride);
  scatter_u(u1_1o,   9,  0, 1, 3,  1, UT, t, stride);
  scatter_u(u1_2e,   9,  0, 1, 5,  4, UT, t, stride);
  // order 2: K rows 9..89, cols 9..31
  scatter_u(u2_0e,  81,  9, 2, 1,  9, UT, t, stride);
  scatter_u(u2_1o,  81,  9, 2, 3, 11, UT, t, stride);
  scatter_u(u2_2e,  81,  9, 3, 5, 17, UT, t, stride);
  // order 3: K rows 90..818, cols 32..93
  scatter_u(u3_0e, 729, 90, 4, 1, 32, UT, t, stride);
  scatter_u(u3_1o, 729, 90, 6, 3, 36, UT, t, stride);
  scatter_u(u3_2e, 729, 90, 8, 5, 54, UT, t, stride);
}

// ---------------- Main fused kernel: mono -> WMMA GEMM -> weighted combine ----
// Block = 192 threads (6 waves). All 6 waves share one 16-row monomial tile in
// LDS; wave w computes column tile w (16 cols) of G = mono(16x820) * U(820x96),
// then 144 threads apply the per-species weight combine and scatter the output.
__global__ __launch_bounds__(BLOCK_THREADS) void symcon_main(
    const float* __restrict__ x, const float* __restrict__ UT,
    const float* __restrict__ w1_0e, const float* __restrict__ w1_1o,
    const float* __restrict__ w1_2e, const float* __restrict__ w2_0e,
    const float* __restrict__ w2_1o, const float* __restrict__ w2_2e,
    const float* __restrict__ w3_0e, const float* __restrict__ w3_1o,
    const float* __restrict__ w3_2e, const int* __restrict__ index,
    float* __restrict__ out) {
  extern __shared__ float lds[];
  float* mono = lds;                                        // [16][820]
  float* G    = lds + TILE_ROWS * UROWS;                    // [16][96]
  float* ldsX = lds + TILE_ROWS * UROWS + TILE_ROWS * UCOLS;// [16*9] staged x rows

  const int tid      = threadIdx.x;
  const int rowBase  = blockIdx.x * TILE_ROWS;  // first (z*C + c) row of tile

  // ---- stage the 16 x-rows (144 contiguous floats) into LDS ----
  if (tid < TILE_ROWS * 9) ldsX[tid] = x[rowBase * 9 + tid];
  __syncthreads();

  // ---- build monomials: 12 threads per row, every 12th entry each ----
  {
    const int row  = tid & 15;
    const int slot = tid >> 4;  // 0..11
    const float* xr = ldsX + row * 9;
    float* mrow = mono + row * UROWS;
    for (int m = slot; m < UROWS; m += 12) {
      float val;
      if (m < 9) {
        val = xr[m];
      } else if (m < 90) {
        int t2 = m - 9;
        int d0 = t2 / 9;
        val = xr[d0] * xr[t2 - d0 * 9];
      } else if (m < 819) {
        int t3 = m - 90;
        int d0 = t3 / 81;
        int rr = t3 - d0 * 81;
        int d1 = rr / 9;
        val = xr[d0] * xr[d1] * xr[rr - d1 * 9];
      } else {
        val = 0.f;  // K padding
      }
      mrow[m] = val;
    }
  }
  __syncthreads();

  // ---- WMMA f32 16x16x4: wave w -> column tile w ----
  // A layout (ISA 7.12.2): lanes 0-15 hold K+0/K+1; lanes 16-31 hold K+2/K+3.
  // B layout mirrors it; UT[col][row] makes each lane's B pair one v2f load.
  {
    // readfirstlane: wave id is physically uniform; move it to an SGPR so the
    // trip count is scalar -> s_cbranch loop the compiler unrolls + pipelines.
    const int wv   = __builtin_amdgcn_readfirstlane(tid >> 5);  // 0..5
    const int lane = tid & 31;
    const int half = lane >> 4;
    const int r    = lane & 15;
    const int ka0  = half ? 2 : 0;
    const int n0   = wv * 16;

    // Per-column-tile K ranges exploiting U's block-zero structure.
    const int kstart[6] = {0, 8, 88, 88, 88, 88};
    const int ksteps[6] = {23, 21, 183, 183, 183, 183};
    const int ns = ksteps[wv];

    const float* arow = mono + r * UROWS;
    const float* bcol = UT + (n0 + r) * UROWS;
    v8f acc = {0.f, 0.f, 0.f, 0.f, 0.f, 0.f, 0.f, 0.f};
    int kb = kstart[wv];
    for (int s = 0; s < ns; ++s, kb += 4) {
      v2f a = *(const v2f*)(arow + kb + ka0);  // LDS, contiguous pair
      v2f b = *(const v2f*)(bcol + kb + ka0);  // global, contiguous pair
      acc = __builtin_amdgcn_wmma_f32_16x16x4_f32(
          /*neg_a=*/false, a, /*neg_b=*/false, b,
          /*c_mod=*/(short)0, acc, /*reuse_a=*/false, /*reuse_b=*/false);
    }
    // C/D layout: VGPR j -> M=j (lanes 0-15) / M=j+8 (lanes 16-31), N=lane&15
#pragma unroll
    for (int j = 0; j < 8; ++j)
      G[(j + 8 * half) * UCOLS + n0 + r] = acc[j];
  }
  __syncthreads();

  // ---- epilogue: one thread per (row, output component) ----
  if (tid < TILE_ROWS * 9) {
    const int row  = tid / 9;
    const int comp = tid - row * 9;
    const int R    = rowBase + row;
    const int z    = R >> 7;
    const int c    = R & 127;
    const int sp   = index[z];
    const float* Gr = G + row * UCOLS;
    float* ob = out + z * 1152;
    if (comp == 0) {
      // 0e (cols: o1=0; o2=9,10; o3=32..35)
      float a = w1_0e[sp * 128 + c] * Gr[0]
              + w2_0e[(sp * 2 + 0) * 128 + c] * Gr[9]
              + w2_0e[(sp * 2 + 1) * 128 + c] * Gr[10];
#pragma unroll
      for (int k = 0; k < 4; ++k) a += w3_0e[(sp * 4 + k) * 128 + c] * Gr[32 + k];
      ob[c] = a;
    } else if (comp < 4) {
      // 1o (cols: o1=1..3; o2=11+3k+i; o3=36+3k+i)
      const int i = comp - 1;
      float a = w1_1o[sp * 128 + c] * Gr[1 + i];
#pragma unroll
      for (int k = 0; k < 2; ++k) a += w2_1o[(sp * 2 + k) * 128 + c] * Gr[11 + 3 * k + i];
#pragma unroll
      for (int k = 0; k < 6; ++k) a += w3_1o[(sp * 6 + k) * 128 + c] * Gr[36 + 3 * k + i];
      ob[128 + c * 3 + i] = a;
    } else {
      // 2e (cols: o1=4..8; o2=17+5k+i; o3=54+5k+i)
      const int i = comp - 4;
      float a = w1_2e[sp * 128 + c] * Gr[4 + i];
#pragma unroll
      for (int k = 0; k < 3; ++k) a += w2_2e[(sp * 3 + k) * 128 + c] * Gr[17 + 5 * k + i];
#pragma unroll
      for (int k = 0; k < 8; ++k) a += w3_2e[(sp * 8 + k) * 128 + c] * Gr[54 + 5 * k + i];
      ob[512 + c * 5 + i] = a;
    }
  }
}

extern "C" void kernel_launch(void* const* d_in, const int* in_sizes, int n_in,
                              void* d_out, int out_size, void* d_ws, size_t ws_size,
                              hipStream_t stream) {
  const float* x     = (const float*)d_in[0];
  const float* u1_0e = (const float*)d_in[1];
  const float* w1_0e = (const float*)d_in[2];
  const float* u1_1o = (const float*)d_in[3];
  const float* w1_1o = (const float*)d_in[4];
  const float* u1_2e = (const float*)d_in[5];
  const float* w1_2e = (const float*)d_in[6];
  const float* u2_0e = (const float*)d_in[7];
  const float* w2_0e = (const float*)d_in[8];
  const float* u2_1o = (const float*)d_in[9];
  const float* w2_1o = (const float*)d_in[10];
  const float* u2_2e = (const float*)d_in[11];
  const float* w2_2e = (const float*)d_in[12];
  const float* u3_0e = (const float*)d_in[13];
  const float* w3_0e = (const float*)d_in[14];
  const float* u3_1o = (const float*)d_in[15];
  const float* w3_1o = (const float*)d_in[16];
  const float* u3_2e = (const float*)d_in[17];
  const float* w3_2e = (const float*)d_in[18];
  const int*   index = (const int*)d_in[19];
  float* UT  = (float*)d_ws;  // 96*820 floats = 315 KB, transposed layout
  float* out = (float*)d_out;

  zero_u<<<(UROWS * UCOLS + 255) / 256, 256, 0, stream>>>(UT);
  fill_u<<<128, 256, 0, stream>>>(u1_0e, u1_1o, u1_2e, u2_0e, u2_1o, u2_2e,
                                  u3_0e, u3_1o, u3_2e, UT);

  const size_t ldsBytes =
      (TILE_ROWS * UROWS + TILE_ROWS * UCOLS + TILE_ROWS * 9) * sizeof(float);
  symcon_main<<<NROW / TILE_ROWS, BLOCK_THREADS, ldsBytes, stream>>>(
      x, UT, w1_0e, w1_1o, w1_2e, w2_0e, w2_1o, w2_2e, w3_0e, w3_1o, w3_2e,
      index, out);
}